// LightGCN_semantic_21930103013809
// MI455X (gfx1250) — compile-verified
//
#include <hip/hip_runtime.h>

typedef __attribute__((ext_vector_type(2))) float v2f;
typedef __attribute__((ext_vector_type(8))) float v8f;

#define D 64
#define Q 16  // float4 chunks per row

__device__ __forceinline__ float dsigmoid(float x) {
    return 1.0f / (1.0f + __expf(-x));
}

// cur = acc = layer-0 fused embeddings
__global__ void fuse0_kernel(const float* __restrict__ ue, const float* __restrict__ ie,
                             const float* __restrict__ se, const float* __restrict__ he,
                             const float* __restrict__ ulw, const float* __restrict__ ilw,
                             float* __restrict__ cur, float* __restrict__ acc,
                             long long U, long long N) {
    long long tid = (long long)blockIdx.x * blockDim.x + threadIdx.x;
    if (tid >= N * Q) return;
    long long node = tid >> 4;
    int q = (int)(tid & 15);
    float4 r;
    if (node < U) {
        float a = dsigmoid(ulw[0]);
        float4 x = reinterpret_cast<const float4*>(ue + node * D)[q];
        float4 y = reinterpret_cast<const float4*>(se + node * D)[q];
        r.x = a * x.x + (1.0f - a) * y.x; r.y = a * x.y + (1.0f - a) * y.y;
        r.z = a * x.z + (1.0f - a) * y.z; r.w = a * x.w + (1.0f - a) * y.w;
    } else {
        float b = dsigmoid(ilw[0]);
        long long it = node - U;
        float4 x = reinterpret_cast<const float4*>(ie + it * D)[q];
        float4 y = reinterpret_cast<const float4*>(he + it * D)[q];
        r.x = b * x.x + (1.0f - b) * y.x; r.y = b * x.y + (1.0f - b) * y.y;
        r.z = b * x.z + (1.0f - b) * y.z; r.w = b * x.w + (1.0f - b) * y.w;
    }
    reinterpret_cast<float4*>(cur + node * D)[q] = r;
    reinterpret_cast<float4*>(acc + node * D)[q] = r;
}

__global__ void zero_kernel(float* __restrict__ p, long long n4) {
    long long tid = (long long)blockIdx.x * blockDim.x + threadIdx.x;
    if (tid >= n4) return;
    reinterpret_cast<float4*>(p)[tid] = make_float4(0.f, 0.f, 0.f, 0.f);
}

// COO SpMM: nxt[row] += val * cur[col]; 16 threads per edge, float4 each.
__global__ void spmm_kernel(const float* __restrict__ cur, float* __restrict__ nxt,
                            const float* __restrict__ ev, const int* __restrict__ er,
                            const int* __restrict__ ec, long long E) {
    long long tid = (long long)blockIdx.x * blockDim.x + threadIdx.x;
    long long e = tid >> 4;
    if (e >= E) return;
    int q = (int)(tid & 15);
    float v = ev[e];
    long long col = (long long)ec[e];
    long long row = (long long)er[e];
    float4 s = reinterpret_cast<const float4*>(cur + col * D)[q];
    float* dst = nxt + row * D + q * 4;
    atomicAdd(dst + 0, v * s.x);
    atomicAdd(dst + 1, v * s.y);
    atomicAdd(dst + 2, v * s.z);
    atomicAdd(dst + 3, v * s.w);
}

// cur = fused(nxt) with layer weights; acc += cur
__global__ void fuse_acc_kernel(const float* __restrict__ nxt,
                                const float* __restrict__ se, const float* __restrict__ he,
                                const float* __restrict__ ulw, const float* __restrict__ ilw,
                                float* __restrict__ cur, float* __restrict__ acc,
                                long long U, long long N, int layer) {
    long long tid = (long long)blockIdx.x * blockDim.x + threadIdx.x;
    if (tid >= N * Q) return;
    long long node = tid >> 4;
    int q = (int)(tid & 15);
    float4 x = reinterpret_cast<const float4*>(nxt + node * D)[q];
    float4 r;
    if (node < U) {
        float a = dsigmoid(ulw[layer]);
        float4 y = reinterpret_cast<const float4*>(se + node * D)[q];
        r.x = a * x.x + (1.0f - a) * y.x; r.y = a * x.y + (1.0f - a) * y.y;
        r.z = a * x.z + (1.0f - a) * y.z; r.w = a * x.w + (1.0f - a) * y.w;
    } else {
        float b = dsigmoid(ilw[layer]);
        long long it = node - U;
        float4 y = reinterpret_cast<const float4*>(he + it * D)[q];
        r.x = b * x.x + (1.0f - b) * y.x; r.y = b * x.y + (1.0f - b) * y.y;
        r.z = b * x.z + (1.0f - b) * y.z; r.w = b * x.w + (1.0f - b) * y.w;
    }
    reinterpret_cast<float4*>(cur + node * D)[q] = r;
    float4 a4 = reinterpret_cast<float4*>(acc + node * D)[q];
    a4.x += r.x; a4.y += r.y; a4.z += r.z; a4.w += r.w;
    reinterpret_cast<float4*>(acc + node * D)[q] = a4;
}

// Batched dot via V_WMMA_F32_16X16X4_F32 diagonal.
// One wave32 computes 16 pairs: C += A(16x4) x B(4x16) over 16 K-chunks (D=64),
// gamma[j] = C[j][j] / 16  (each side is acc/4).
// A layout (32-bit 16x4): lane m=l%16; VGPR0 = K = 2*(l/16); VGPR1 = K+1.
// B layout (4x16, row striped): same indexing with the item row — symmetric.
// C/D diag: j<8 at lane j, vgpr j; j>=8 at lane j+16, vgpr j-8.
__global__ void dot16_wmma_kernel(const float* __restrict__ acc,
                                  const int* __restrict__ users,
                                  const int* __restrict__ items,
                                  float* __restrict__ out,
                                  long long U, int B) {
    int wave = (blockIdx.x * blockDim.x + threadIdx.x) >> 5;
    int lane = threadIdx.x & 31;
    int base = wave * 16;
    if (base >= B) return;

    int m  = lane & 15;
    int hi = lane >> 4;          // 0: K even slot, 1: odd-pair slot

    long long urow = (long long)users[base + m];
    long long irow = (long long)(U + items[base + m]);
    const float* up = acc + urow * D;
    const float* ip = acc + irow * D;

    v8f c = {0.f, 0.f, 0.f, 0.f, 0.f, 0.f, 0.f, 0.f};
#pragma unroll
    for (int kc = 0; kc < 16; ++kc) {
        int k0 = 4 * kc + 2 * hi;      // VGPR0 holds K=k0, VGPR1 holds K=k0+1
        v2f a, b;
        a.x = up[k0]; a.y = up[k0 + 1];
        b.x = ip[k0]; b.y = ip[k0 + 1];
        c = __builtin_amdgcn_wmma_f32_16x16x4_f32(
                /*neg_a=*/false, a, /*neg_b=*/false, b,
                /*c_mod=*/(short)0, c, /*reuse_a=*/false, /*reuse_b=*/false);
    }

    // extract diagonal
    bool lo = (lane < 8);
    bool hi8 = (lane >= 24);
    if (lo || hi8) {
        int idx = lo ? lane : (lane - 24);   // which VGPR of C holds my diag
        int j   = lo ? lane : (lane - 16);   // which output element
        float d = 0.f;
#pragma unroll
        for (int r = 0; r < 8; ++r)
            if (idx == r) d = c[r];
        out[base + j] = d * 0.0625f;         // (acc/4)·(acc/4)
    }
}

extern "C" void kernel_launch(void* const* d_in, const int* in_sizes, int n_in,
                              void* d_out, int out_size, void* d_ws, size_t ws_size,
                              hipStream_t stream) {
    const float* user_emb    = (const float*)d_in[0];
    const float* item_emb    = (const float*)d_in[1];
    const float* symptom_emb = (const float*)d_in[2];
    const float* herb_emb    = (const float*)d_in[3];
    const float* user_lw     = (const float*)d_in[4];
    const float* item_lw     = (const float*)d_in[5];
    const float* edge_val    = (const float*)d_in[6];
    const int*   edge_row    = (const int*)d_in[7];
    const int*   edge_col    = (const int*)d_in[8];
    const int*   users       = (const int*)d_in[9];
    const int*   items       = (const int*)d_in[10];
    float* out = (float*)d_out;

    const long long U = in_sizes[0] / D;
    const long long I = in_sizes[1] / D;
    const long long N = U + I;
    const long long E = in_sizes[6];
    const int       B = in_sizes[9];
    const int       L = 3;

    float* cur = (float*)d_ws;
    float* nxt = cur + N * D;
    float* acc = nxt + N * D;

    const int TB = 256;
    long long fuse_threads = N * Q;
    int fuse_blocks = (int)((fuse_threads + TB - 1) / TB);
    long long spmm_threads = E * Q;
    int spmm_blocks = (int)((spmm_threads + TB - 1) / TB);

    fuse0_kernel<<<fuse_blocks, TB, 0, stream>>>(user_emb, item_emb, symptom_emb, herb_emb,
                                                 user_lw, item_lw, cur, acc, U, N);
    for (int layer = 1; layer <= L; ++layer) {
        zero_kernel<<<fuse_blocks, TB, 0, stream>>>(nxt, N * Q);
        spmm_kernel<<<spmm_blocks, TB, 0, stream>>>(cur, nxt, edge_val, edge_row, edge_col, E);
        fuse_acc_kernel<<<fuse_blocks, TB, 0, stream>>>(nxt, symptom_emb, herb_emb,
                                                        user_lw, item_lw, cur, acc,
                                                        U, N, layer);
    }
    int dot_waves = (B + 15) / 16;                 // 1024
    int dot_blocks = (dot_waves * 32 + TB - 1) / TB;
    dot16_wmma_kernel<<<dot_blocks, TB, 0, stream>>>(acc, users, items, out, U, B);
}